// DynamicConvolutionAttention_12850542149784
// MI455X (gfx1250) — compile-verified
//
#include <hip/hip_runtime.h>
#include <hip/hip_bf16.h>
#include <math.h>

// ---------------- problem constants ----------------
#define BB   64
#define TT   2048
#define QD   1024
#define MD   512
#define AD   128
#define SCN  8
#define SKN  21
#define DCN  8
#define DKN  21
#define GW   (DCN * DKN)   // 168
#define PLEN 11

typedef __attribute__((ext_vector_type(2))) float v2f;
typedef __attribute__((ext_vector_type(8))) float v8f;

// =====================================================================
// Kernel 1: h[B,AD] = tanh(query @ Wq^T + bq)
// WMMA f32 16x16x4. One wave per 16x16 output tile, K=QD in steps of 4.
// A[m,k]=query[m,k], B[k,n]=Wq[n,k].
// Fragment layout (wave32): lanes 0-15 -> m(n)=lane, k = k0,k0+1
//                           lanes 16-31 -> m(n)=lane-16, k = k0+2,k0+3
// =====================================================================
__global__ __launch_bounds__(32)
void k1_h_wmma(const float* __restrict__ query,
               const float* __restrict__ Wq,
               const float* __restrict__ bq,
               float* __restrict__ h)
{
    const int lane = threadIdx.x;
    const int mtile = blockIdx.x >> 3;   // 0..3   (64/16)
    const int ntile = blockIdx.x & 7;    // 0..7   (128/16)
    const int m0 = mtile * 16;
    const int n0 = ntile * 16;

    const int half = lane >> 4;          // 0 | 1
    const int ml   = lane & 15;

    const float* arow = query + (size_t)(m0 + ml) * QD;
    const float* brow = Wq    + (size_t)(n0 + ml) * QD;

    v8f c = {};
    #pragma unroll 4
    for (int k0 = 0; k0 < QD; k0 += 4) {
        const int ka = k0 + half * 2;
        v2f a = *(const v2f*)(arow + ka);
        v2f b = *(const v2f*)(brow + ka);
        c = __builtin_amdgcn_wmma_f32_16x16x4_f32(false, a, false, b,
                                                  (short)0, c, false, false);
    }

    // C layout: VGPR r, lanes 0-15 -> (m0+r, n0+lane) ; lanes 16-31 -> (m0+r+8, n0+lane-16)
    const float bqv = bq[n0 + ml];
    #pragma unroll
    for (int r = 0; r < 8; ++r) {
        const int m = m0 + r + half * 8;
        h[(size_t)m * AD + (n0 + ml)] = tanhf(c[r] + bqv);
    }
}

// =====================================================================
// Kernel 2: G[B,168] = h @ Wk^T   (tiny, VALU, float4 loads)
// =====================================================================
__global__ __launch_bounds__(256)
void k2_G(const float* __restrict__ h,
          const float* __restrict__ Wk,
          float* __restrict__ G)
{
    const int idx = blockIdx.x * blockDim.x + threadIdx.x; // 64*168 = 10752
    if (idx >= BB * GW) return;
    const int b = idx / GW;
    const int j = idx - b * GW;
    const float4* hv = (const float4*)(h  + (size_t)b * AD);
    const float4* wv = (const float4*)(Wk + (size_t)j * AD);
    float acc = 0.f;
    #pragma unroll 8
    for (int a = 0; a < AD / 4; ++a) {
        float4 x = hv[a], w = wv[a];
        acc += x.x * w.x + x.y * w.y + x.z * w.z + x.w * w.w;
    }
    G[idx] = acc;
}

// =====================================================================
// Kernel 3: alignment[b,t] = Wv . tanh(F @ W2^T + bdl) + log(clip(prior*aw))
// One wave handles 16 consecutive t positions of one batch row.
// F (16 pos x 16 feat) built in LDS, then 8 N-tiles x 4 K-steps of WMMA f32.
// =====================================================================
__global__ __launch_bounds__(32)
void k3_energy(const float* __restrict__ aw,
               const float* __restrict__ prior,
               const float* __restrict__ G,
               const float* __restrict__ Wsf,
               const float* __restrict__ Wsl,
               const float* __restrict__ Wdl,
               const float* __restrict__ bdl,
               const float* __restrict__ Wv,
               float* __restrict__ alignment)
{
    __shared__ float awwin[36];
    __shared__ float F[16][17];        // +1 pad vs bank conflicts
    __shared__ float part[32][8];

    const int lane = threadIdx.x;
    const int b  = blockIdx.x >> 7;          // / (TT/16)
    const int t0 = (blockIdx.x & 127) * 16;

    // stage aw window [t0-10, t0+25] with boundary zeros
    for (int j = lane; j < 36; j += 32) {
        const int tg = t0 - (DKN - 1) / 2 + j;
        awwin[j] = (tg >= 0 && tg < TT) ? aw[(size_t)b * TT + tg] : 0.f;
    }
    __syncthreads();

    // features: lanes 0-15 -> static filters, lanes 16-31 -> dynamic filters
    {
        const int pos = lane & 15;
        if (lane < 16) {
            #pragma unroll
            for (int s = 0; s < SCN; ++s) {
                float acc = 0.f;
                #pragma unroll
                for (int k = 0; k < SKN; ++k) acc += awwin[pos + k] * Wsf[s * SKN + k];
                F[pos][s] = acc;
            }
        } else {
            const float* g = G + (size_t)b * GW;
            #pragma unroll
            for (int c = 0; c < DCN; ++c) {
                float acc = 0.f;
                #pragma unroll
                for (int k = 0; k < DKN; ++k) acc += awwin[pos + k] * g[c * DKN + k];
                F[pos][SCN + c] = acc;
            }
        }
    }
    __syncthreads();

    const int half = lane >> 4;
    const int ml   = lane & 15;

    // A fragments from LDS (k pattern: half 0 -> k0,k0+1 ; half 1 -> k0+2,k0+3)
    v2f afrag[4];
    #pragma unroll
    for (int kk = 0; kk < 4; ++kk) {
        const int k0 = kk * 4 + half * 2;
        afrag[kk].x = F[ml][k0];
        afrag[kk].y = F[ml][k0 + 1];
    }

    float epart[8];
    #pragma unroll
    for (int r = 0; r < 8; ++r) epart[r] = 0.f;

    // hidden = F @ W2^T, W2[n][i<8]=Wsl[n][i], W2[n][i>=8]=Wdl[n][i-8]
    for (int n0 = 0; n0 < AD; n0 += 16) {
        const int n = n0 + ml;
        v8f c = {};
        #pragma unroll
        for (int kk = 0; kk < 4; ++kk) {
            const int k0 = kk * 4 + half * 2;   // even; k0 and k0+1 in same matrix
            v2f bf;
            if (k0 < SCN) {
                bf.x = Wsl[n * SCN + k0];
                bf.y = Wsl[n * SCN + k0 + 1];
            } else {
                bf.x = Wdl[n * DCN + k0 - SCN];
                bf.y = Wdl[n * DCN + k0 + 1 - SCN];
            }
            c = __builtin_amdgcn_wmma_f32_16x16x4_f32(false, afrag[kk], false, bf,
                                                      (short)0, c, false, false);
        }
        const float wv = Wv[n];
        const float bb = bdl[n];
        #pragma unroll
        for (int r = 0; r < 8; ++r)
            epart[r] += tanhf(c[r] + bb) * wv;   // (m = r + 8*half, n-slice = n)
    }

    #pragma unroll
    for (int r = 0; r < 8; ++r) part[lane][r] = epart[r];
    __syncthreads();

    if (lane < 16) {
        const int m = lane;
        float e = 0.f;
        if (m < 8) {
            #pragma unroll
            for (int l = 0; l < 16; ++l) e += part[l][m];
        } else {
            #pragma unroll
            for (int l = 16; l < 32; ++l) e += part[l][m - 8];
        }
        // causal prior filter + log
        float pf = 0.f;
        #pragma unroll
        for (int p = 0; p < PLEN; ++p) pf += awwin[m + p] * prior[p];
        const float pl = logf(fmaxf(pf, 1e-6f));
        alignment[(size_t)b * TT + t0 + m] = e + pl;
    }
}

// =====================================================================
// Kernel 4: softmax over T per batch row + mask
// =====================================================================
__global__ __launch_bounds__(256)
void k4_softmax(const float* __restrict__ alignment,
                const unsigned char* __restrict__ mask,
                float* __restrict__ awsoft)
{
    __shared__ float red[256];
    const int b = blockIdx.x;
    const int tid = threadIdx.x;
    const float* row = alignment + (size_t)b * TT;

    float lmax = -3.4e38f;
    for (int t = tid; t < TT; t += 256) lmax = fmaxf(lmax, row[t]);
    red[tid] = lmax; __syncthreads();
    for (int s = 128; s > 0; s >>= 1) {
        if (tid < s) red[tid] = fmaxf(red[tid], red[tid + s]);
        __syncthreads();
    }
    const float mx = red[0]; __syncthreads();

    float lsum = 0.f;
    for (int t = tid; t < TT; t += 256) lsum += __expf(row[t] - mx);
    red[tid] = lsum; __syncthreads();
    for (int s = 128; s > 0; s >>= 1) {
        if (tid < s) red[tid] += red[tid + s];
        __syncthreads();
    }
    const float inv = 1.f / red[0];

    for (int t = tid; t < TT; t += 256) {
        const float v = __expf(row[t] - mx) * inv;
        awsoft[(size_t)b * TT + t] = mask[(size_t)b * TT + t] ? v : 1e-8f;
    }
}

// =====================================================================
// Kernel 5a: zero output (d_out is poisoned before each timing run)
// =====================================================================
__global__ __launch_bounds__(256)
void k5a_zero(float* __restrict__ out, int n)
{
    const int i = blockIdx.x * blockDim.x + threadIdx.x;
    if (i < n) out[i] = 0.f;
}

// =====================================================================
// Kernel 5b: context[b,d] = sum_t aw[b,t] * memory[b,t,d]
// grid (tchunk=8, dblk=4, b=64), 256 threads. Softmax weights staged into
// LDS via CDNA5 async-to-LDS (ASYNCcnt path), memory streamed with b128
// loads + prefetch, LDS partial reduce, f32 global atomics for the T split.
// =====================================================================
__global__ __launch_bounds__(256)
void k5b_context(const float* __restrict__ awsoft,
                 const float* __restrict__ memory,
                 float* __restrict__ out)
{
    __shared__ float aw_s[256];
    __shared__ float partial[8][32][4];

    const int b    = blockIdx.z;
    const int dblk = blockIdx.y;   // 0..3  -> 128 floats each
    const int tch  = blockIdx.x;   // 0..7  -> 256 t each
    const int tid  = threadIdx.x;
    const int tg   = tid >> 5;     // 0..7
    const int dl   = tid & 31;     // 0..31

    const int tbase = tch * 256;

#if defined(__gfx1250__)
    // CDNA5 async copy: global -> LDS without a VGPR round-trip.
    // Flat LDS pointers truncate to the LDS byte offset (addr[31:0]).
    {
        const float* gsrc = awsoft + (size_t)b * TT + tbase + tid;
        unsigned ldsa = (unsigned)(uintptr_t)(&aw_s[tid]);
        asm volatile("global_load_async_to_lds_b32 %0, %1, off"
                     :: "v"(ldsa), "v"(gsrc)
                     : "memory");
        asm volatile("s_wait_asynccnt 0x0" ::: "memory");
    }
#else
    aw_s[tid] = awsoft[(size_t)b * TT + tbase + tid];
#endif
    __syncthreads();

    const int d0 = dblk * 128 + dl * 4;
    const float* memb = memory + ((size_t)b * TT) * MD + d0;

    float4 acc = make_float4(0.f, 0.f, 0.f, 0.f);
    for (int tl = tg; tl < 256; tl += 8) {
        const float w = aw_s[tl];
        const float* p = memb + (size_t)(tbase + tl) * MD;
        __builtin_prefetch(p + 8 * MD, 0, 0);           // global_prefetch_b8
        const float4 m = *reinterpret_cast<const float4*>(p);
        acc.x += w * m.x; acc.y += w * m.y; acc.z += w * m.z; acc.w += w * m.w;
    }
    partial[tg][dl][0] = acc.x;
    partial[tg][dl][1] = acc.y;
    partial[tg][dl][2] = acc.z;
    partial[tg][dl][3] = acc.w;
    __syncthreads();

    if (tg == 0) {
        float s0 = 0.f, s1 = 0.f, s2 = 0.f, s3 = 0.f;
        #pragma unroll
        for (int g = 0; g < 8; ++g) {
            s0 += partial[g][dl][0];
            s1 += partial[g][dl][1];
            s2 += partial[g][dl][2];
            s3 += partial[g][dl][3];
        }
        float* o = out + (size_t)b * MD + d0;
        atomicAdd(o + 0, s0);
        atomicAdd(o + 1, s1);
        atomicAdd(o + 2, s2);
        atomicAdd(o + 3, s3);
    }
}

// =====================================================================
// Host-side launcher
// =====================================================================
extern "C" void kernel_launch(void* const* d_in, const int* in_sizes, int n_in,
                              void* d_out, int out_size, void* d_ws, size_t ws_size,
                              hipStream_t stream)
{
    const float*         query  = (const float*)d_in[0];
    const float*         memory = (const float*)d_in[1];
    const unsigned char* mask   = (const unsigned char*)d_in[2];
    const float*         aw     = (const float*)d_in[3];
    const float*         prior  = (const float*)d_in[4];
    const float*         Wq     = (const float*)d_in[5];
    const float*         bq     = (const float*)d_in[6];
    const float*         Wk     = (const float*)d_in[7];
    const float*         Wsf    = (const float*)d_in[8];
    const float*         Wsl    = (const float*)d_in[9];
    const float*         Wdl    = (const float*)d_in[10];
    const float*         bdl    = (const float*)d_in[11];
    const float*         Wv     = (const float*)d_in[12];
    float*               out    = (float*)d_out;

    // workspace layout (floats)
    float* h      = (float*)d_ws;                 // [64,128]
    float* G      = h + BB * AD;                  // [64,168]
    float* align_ = G + BB * GW;                  // [64,2048]
    float* awsoft = align_ + (size_t)BB * TT;     // [64,2048]

    // zero the output early (independent of the pipeline)
    k5a_zero<<<(BB * MD + 255) / 256, 256, 0, stream>>>(out, BB * MD);

    // 1) h = tanh(q Wq^T + bq)   -- WMMA f32 16x16x4
    k1_h_wmma<<<32, 32, 0, stream>>>(query, Wq, bq, h);

    // 2) G = h Wk^T
    k2_G<<<(BB * GW + 255) / 256, 256, 0, stream>>>(h, Wk, G);

    // 3) alignment = energy + log-prior   -- WMMA f32 16x16x4 per 16-pos tile
    k3_energy<<<BB * (TT / 16), 32, 0, stream>>>(aw, prior, G, Wsf, Wsl, Wdl,
                                                 bdl, Wv, align_);

    // 4) softmax + mask
    k4_softmax<<<BB, 256, 0, stream>>>(align_, mask, awsoft);

    // 5) context = aw . memory  (HBM-bound: 256 MB @ 23.3 TB/s ~ 11 us)
    dim3 grid5(8, 4, BB);
    k5b_context<<<grid5, 256, 0, stream>>>(awsoft, memory, out);
}